// OTNorm_29669634081222
// MI455X (gfx1250) — compile-verified
//
#include <hip/hip_runtime.h>

#define B_  64
#define T_  4096
#define C_  128
#define Q_  64
#define EPS_ 1e-8f

typedef __attribute__((ext_vector_type(4))) unsigned int u32x4;
typedef __attribute__((ext_vector_type(8))) unsigned int u32x8;

// Single-precision erfinv (M. Giles, 2010 approximation)
__device__ __forceinline__ float erfinv_poly(float y) {
    float w = -__logf(fmaxf((1.0f - y) * (1.0f + y), 1e-30f));
    float p;
    if (w < 5.0f) {
        w -= 2.5f;
        p =                2.81022636e-08f;
        p = fmaf(p, w,     3.43273939e-07f);
        p = fmaf(p, w,    -3.5233877e-06f);
        p = fmaf(p, w,    -4.39150654e-06f);
        p = fmaf(p, w,     0.00021858087f);
        p = fmaf(p, w,    -0.00125372503f);
        p = fmaf(p, w,    -0.00417768164f);
        p = fmaf(p, w,     0.246640727f);
        p = fmaf(p, w,     1.50140941f);
    } else {
        w = sqrtf(w) - 3.0f;
        p =               -0.000200214257f;
        p = fmaf(p, w,     0.000100950558f);
        p = fmaf(p, w,     0.00134934322f);
        p = fmaf(p, w,    -0.00367342844f);
        p = fmaf(p, w,     0.00573950773f);
        p = fmaf(p, w,    -0.0076224613f);
        p = fmaf(p, w,     0.00943887047f);
        p = fmaf(p, w,     1.00167406f);
        p = fmaf(p, w,     2.83297682f);
    }
    return p * y;
}

// Build a D# for a 1-wide x T_-tall tile over a tensor with dim0 stride C_
// (i.e. one time-row of a (B,T,C) array at fixed b,c). Direction (load vs
// store) is selected by the opcode, not the descriptor, for user-mode D#.
__device__ __forceinline__ void tdm_row_desc(u32x4& g0, u32x8& g1,
                                             unsigned long long ga,
                                             unsigned lds_off) {
    g0[0] = __builtin_amdgcn_readfirstlane(1u);                    // count=1 (valid)
    g0[1] = __builtin_amdgcn_readfirstlane(lds_off);               // lds_addr (bytes)
    g0[2] = __builtin_amdgcn_readfirstlane((unsigned)ga);          // global_addr[31:0]
    g0[3] = __builtin_amdgcn_readfirstlane(
                ((unsigned)(ga >> 32) & 0x1FFFFFFu) | 0x80000000u); // global_addr[56:32] | type=2

    g1[0] = __builtin_amdgcn_readfirstlane(2u << 16);              // wg_mask=0, data_size=2 (4B)
    g1[1] = __builtin_amdgcn_readfirstlane(1u << 16);              // tensor_dim0[15:0]=1    @ [63:48]
    g1[2] = __builtin_amdgcn_readfirstlane((unsigned)T_ << 16);    // tensor_dim1[15:0]=4096 @ [95:80]
    g1[3] = __builtin_amdgcn_readfirstlane(1u << 16);              // tile_dim0=1            @ [127:112]
    g1[4] = __builtin_amdgcn_readfirstlane((unsigned)T_);          // tile_dim1=4096, tile_dim2=0
    g1[5] = __builtin_amdgcn_readfirstlane((unsigned)C_);          // tensor_dim0_stride = C_ elements
    g1[6] = __builtin_amdgcn_readfirstlane(0u);
    g1[7] = __builtin_amdgcn_readfirstlane(0u);
}

__global__ __launch_bounds__(256) void otnorm_kernel(const float* __restrict__ x,
                                                     float* __restrict__ out) {
    __shared__ float s_data[T_];   // one (b,c) row, time-contiguous; reused for output z
    __shared__ float s_q[Q_];      // quantile table

    const int tid      = threadIdx.x;
    const unsigned row = blockIdx.x;
    const unsigned b   = row >> 7;        // C_ == 128
    const unsigned c   = row & (C_ - 1);

    const unsigned long long elem_off =
        (unsigned long long)b * (unsigned long long)(T_ * C_) + c;
    const unsigned lds_off = (unsigned)(uintptr_t)&s_data[0]; // low 32 generic bits = LDS offset

    // ---- TDM: DMA the strided row x[b, :, c] into LDS ----
    if (tid < 32) {  // wave 0 issues the tensor op (TENSORcnt is per-wave)
        u32x4 g0; u32x8 g1;
        tdm_row_desc(g0, g1, (unsigned long long)(uintptr_t)x + elem_off * 4ull, lds_off);
        asm volatile("tensor_load_to_lds %0, %1" :: "s"(g0), "s"(g1) : "memory");
        __builtin_amdgcn_s_wait_tensorcnt(0);
    }
    __syncthreads();   // publish TDM-written LDS to all 8 waves

    // ---- snapshot original values (t = tid + 256k): conflict-free stride-256 reads ----
    float xv[16];
#pragma unroll
    for (int k = 0; k < 16; ++k) xv[k] = s_data[tid + k * 256];

    // ---- bitonic sort of 4096 floats in LDS (ascending) ----
    for (int k = 2; k <= T_; k <<= 1) {
        for (int j = k >> 1; j > 0; j >>= 1) {
            __syncthreads();
            for (int t = tid; t < T_ / 2; t += 256) {
                int i = 2 * t - (t & (j - 1));   // low index of pair
                int p = i + j;                   // partner
                float a  = s_data[i];
                float bb = s_data[p];
                bool up = ((i & k) == 0);
                if ((a > bb) == up) { s_data[i] = bb; s_data[p] = a; }
            }
        }
    }
    __syncthreads();

    // ---- quantiles: positions are exactly 64i+31.5 -> mean of two order stats ----
    if (tid < Q_) s_q[tid] = 0.5f * (s_data[tid * 64 + 31] + s_data[tid * 64 + 32]);
    __syncthreads();   // everyone sees s_q; s_data is now dead -> reuse for z

    // ---- CDF interp + probit; write z back into s_data[t] ----
    const float invQ = 1.0f / (float)Q_;
#pragma unroll
    for (int k = 0; k < 16; ++k) {
        float xx = xv[k];
        // branchless binary search: pos = #(q <= xx) (caps at 63 by construction)
        int pos = 0;
#pragma unroll
        for (int s = 32; s >= 1; s >>= 1) {
            int np = pos + s;                 // np <= 63 always
            pos = (s_q[np - 1] <= xx) ? np : pos;
        }
        int idx = (pos < 1) ? 1 : pos;        // clip to [1, Q-1]
        float x0 = s_q[idx - 1];
        float x1 = s_q[idx];
        float y0 = ((float)idx - 0.5f) * invQ;           // levels[idx-1]
        float slope = invQ / fmaxf(x1 - x0, 1e-12f);     // levels are uniform: dy = 1/Q
        float pcdf = fmaf(slope, xx - x0, y0);
        pcdf = fminf(fmaxf(pcdf, EPS_), 1.0f - EPS_);
        float z = erfinv_poly(2.0f * pcdf - 1.0f) * 1.41421356237309505f;

        s_data[tid + k * 256] = z;            // stage result for TDM scatter
    }
    __syncthreads();   // all z values staged in LDS

    // ---- TDM: DMA-scatter z row from LDS to out[b, :, c] (mirror descriptor) ----
    if (tid < 32) {
        u32x4 g0; u32x8 g1;
        tdm_row_desc(g0, g1, (unsigned long long)(uintptr_t)out + elem_off * 4ull, lds_off);
        asm volatile("tensor_store_from_lds %0, %1" :: "s"(g0), "s"(g1) : "memory");
        __builtin_amdgcn_s_wait_tensorcnt(0);  // DMA's LDS reads done before WG teardown
    }
}

extern "C" void kernel_launch(void* const* d_in, const int* in_sizes, int n_in,
                              void* d_out, int out_size, void* d_ws, size_t ws_size,
                              hipStream_t stream) {
    const float* x = (const float*)d_in[0];
    float* out = (float*)d_out;
    (void)in_sizes; (void)n_in; (void)out_size; (void)d_ws; (void)ws_size;
    otnorm_kernel<<<dim3(B_ * C_), dim3(256), 0, stream>>>(x, out);
}